// Detector_50714973831525
// MI455X (gfx1250) — compile-verified
//
#include <hip/hip_runtime.h>
#include <stdint.h>

// ---------------------------------------------------------------------------
// Greedy peak-IoU NMS for MI455X (gfx1250, wave32).
//
// Tiny, sequentially-dependent problem (320 KB in / 256 KB out): the CDNA5
// play is to run the whole pipeline on one WGP out of the 320 KB LDS.
//  - 128 KB in-LDS bitonic sort of (conf,idx) keys
//  - 262 KB LDS-resident pivot NMS with ds_or_b32 suppression bitmasks
//  - global->LDS fills use GLOBAL_LOAD_ASYNC_TO_LDS_B128 (ASYNCcnt path,
//    no VGPR round-trip) + s_wait_asynccnt
// WMMA does not apply: peak-IoU is min/max/div with data-dependent control,
// not a contraction, and precision-lowering would break f32-exact thresholds.
// ---------------------------------------------------------------------------

#define NBOX   16384
#define NWORDS (NBOX / 32)          // 512 words of bitmask
#define CONF_THRESH 0.5f
#define PIOU_THRESH 0.5f

// LDS byte offset of a __shared__ object: the low 32 bits of its LDS-aperture
// flat address are the wave-relative LDS offset (CDNA5 ISA).
__device__ __forceinline__ unsigned lds_off(const void* p) {
    return (unsigned)(uintptr_t)p;
}

// Async copy: 16 bytes from global `g` into LDS offset `loff` (per lane).
// GV mode: VDST = LDS byte-address VGPR, VADDR = 64-bit global address.
__device__ __forceinline__ void async_ld_b128(unsigned loff, const void* g) {
    asm volatile("global_load_async_to_lds_b128 %0, %1, off"
                 :: "v"(loff), "v"(g) : "memory");
}
__device__ __forceinline__ void wait_asynccnt0() {
    asm volatile("s_wait_asynccnt 0x0" ::: "memory");
}

// ---------------------------------------------------------------------------
// 1) Pack sort keys: descending sort of ((conf_bits<<32) | (0xFFFFFFFF - i))
//    == argsort(-conf) with stable tie-break (smaller original index first).
//    conf in [0,1) is non-negative -> IEEE bits are order-preserving.
//    Also zero the valid-bitmask words (gather OR's into them later).
// ---------------------------------------------------------------------------
__global__ void pack_kernel(const float* __restrict__ out5,
                            unsigned long long* __restrict__ items,
                            unsigned* __restrict__ validmask) {
    int i = blockIdx.x * blockDim.x + threadIdx.x;
    if (i < NWORDS) validmask[i] = 0u;
    if (i < NBOX) {
        float conf = out5[(size_t)i * 5 + 0];
        unsigned cb = __float_as_uint(conf);
        items[i] = ((unsigned long long)cb << 32)
                 | (unsigned long long)(0xFFFFFFFFu - (unsigned)i);
    }
}

// ---------------------------------------------------------------------------
// 2) Full bitonic sort (descending) of 16384 uint64 keys in 128 KB static LDS.
//    One workgroup (32 wave32 waves); fill via async-to-LDS B128.
// ---------------------------------------------------------------------------
__global__ void __launch_bounds__(1024)
sort_kernel(unsigned long long* __restrict__ items) {
    __shared__ alignas(16) unsigned long long s[NBOX];   // 128 KB
    const int nt  = blockDim.x;
    const int tid = threadIdx.x;

    // async bulk fill: 2 uint64 (16 B) per lane per op, 8 passes
    for (int base = 0; base < NBOX; base += nt * 2) {
        int e = base + tid * 2;
        async_ld_b128(lds_off(&s[e]), items + e);
    }
    wait_asynccnt0();
    __syncthreads();

    for (unsigned k = 2; k <= NBOX; k <<= 1) {
        for (unsigned j = k >> 1; j > 0; j >>= 1) {
            for (unsigned i = tid; i < NBOX; i += nt) {
                unsigned l = i ^ j;
                if (l > i) {
                    unsigned long long a = s[i], b = s[l];
                    // (i & k)==0 segments sort DESCENDING -> overall descending
                    bool asc = (i & k) != 0;
                    bool sw  = asc ? (a > b) : (a < b);
                    if (sw) { s[i] = b; s[l] = a; }
                }
            }
            __syncthreads();
        }
    }
    for (int i = tid; i < NBOX; i += nt) items[i] = s[i];
}

// ---------------------------------------------------------------------------
// 3) Gather sorted boxes into SoA arrays + build the valid bitmask.
// ---------------------------------------------------------------------------
__global__ void gather_kernel(const float* __restrict__ out5,
                              const unsigned long long* __restrict__ items,
                              float* __restrict__ st, float* __restrict__ en,
                              float* __restrict__ pk, float* __restrict__ hh,
                              unsigned* __restrict__ validmask) {
    int i = blockIdx.x * blockDim.x + threadIdx.x;
    if (i >= NBOX) return;
    unsigned long long it = items[i];
    unsigned orig = 0xFFFFFFFFu - (unsigned)(it & 0xFFFFFFFFull);
    const float* b = out5 + (size_t)orig * 5;
    __builtin_prefetch(b, 0, 0);                 // global_prefetch_b8
    float conf = __uint_as_float((unsigned)(it >> 32));
    st[i] = b[1]; en[i] = b[2]; pk[i] = b[3]; hh[i] = b[4];
    if (conf > CONF_THRESH) atomicOr(&validmask[i >> 5], 1u << (i & 31));
}

// ---------------------------------------------------------------------------
// 4) Pivot-driven NMS, fully LDS-resident (256 KB box data + 6 KB bitmasks).
//    Thread 0 serially finds the next surviving pivot (amortized O(N) total);
//    all 1024 lanes suppress later boxes in parallel with exact f32 peak-IoU.
// ---------------------------------------------------------------------------
__global__ void __launch_bounds__(1024)
nms_kernel(const float* __restrict__ gst, const float* __restrict__ gen,
           const float* __restrict__ gpk, const float* __restrict__ ghh,
           const unsigned* __restrict__ validmask,
           unsigned* __restrict__ keepmask_out) {
    __shared__ alignas(16) float s_st[NBOX];             // 4 x 64 KB = 256 KB
    __shared__ alignas(16) float s_en[NBOX];
    __shared__ alignas(16) float s_pk[NBOX];
    __shared__ alignas(16) float s_h [NBOX];
    __shared__ unsigned s_valid[NWORDS], s_sup[NWORDS], s_keep[NWORDS];
    __shared__ int s_piv;
    const int nt  = blockDim.x;
    const int tid = threadIdx.x;

    // async bulk fill: 4 floats (16 B) per lane per op, 4 passes per array
    {
        const float* gsrc[4] = {gst, gen, gpk, ghh};
        float*       ldst[4] = {s_st, s_en, s_pk, s_h};
        for (int a = 0; a < 4; ++a) {
            for (int base = 0; base < NBOX; base += nt * 4) {
                int e = base + tid * 4;
                async_ld_b128(lds_off(&ldst[a][e]), gsrc[a] + e);
            }
        }
    }
    for (int w = tid; w < NWORDS; w += nt) {
        s_valid[w] = validmask[w]; s_sup[w] = 0u; s_keep[w] = 0u;
    }
    wait_asynccnt0();
    __syncthreads();

    int cur = 0;
    for (;;) {
        if (tid == 0) {
            int p = -1;
            for (int i = cur; i < NBOX; ++i) {
                unsigned live = s_valid[i >> 5] & ~s_sup[i >> 5];
                if (live & (1u << (i & 31))) { p = i; break; }
            }
            s_piv = p;
            if (p >= 0) s_keep[p >> 5] |= (1u << (p & 31));
        }
        __syncthreads();
        const int p = s_piv;            // uniform across the workgroup
        if (p < 0) break;
        cur = p + 1;

        const float pst = s_st[p], pen = s_en[p], ppk = s_pk[p], ph = s_h[p];
        const float parea = (pen - pst) * ph;

        for (int j = p + 1 + tid; j < NBOX; j += nt) {
            const unsigned w = (unsigned)j >> 5, bm = 1u << (j & 31);
            if (s_sup[w] & bm) continue;          // already dead
            const float bst = s_st[j], ben = s_en[j];
            const float bpk = s_pk[j], bh = s_h[j];
            // exact f32 mirror of _peak_iou_row
            const float is    = fmaxf(pst, bst);
            const float ie    = fminf(pen, ben);
            const float il    = fmaxf(ie - is, 0.0f);
            const float ih    = fminf(ph, bh);
            const float inter = il * ih;
            const float uni   = parea + (ben - bst) * bh - inter;
            const float iou   = inter / uni;
            const float ud    = fabsf(fmaxf(pen, ben) - fminf(pst, bst));
            const float piou  = iou - fabsf(ppk - bpk) / ud;
            if (piou > PIOU_THRESH) atomicOr(&s_sup[w], bm);  // ds_or_b32
        }
        __syncthreads();
    }

    for (int w = tid; w < NWORDS; w += nt) keepmask_out[w] = s_keep[w];
}

// ---------------------------------------------------------------------------
// 5) Emit [N,4] = sorted [start,end,peak,height] * keep (one b128 per box).
// ---------------------------------------------------------------------------
__global__ void write_kernel(const float* __restrict__ st, const float* __restrict__ en,
                             const float* __restrict__ pk, const float* __restrict__ hh,
                             const unsigned* __restrict__ keepmask,
                             float* __restrict__ out) {
    int i = blockIdx.x * blockDim.x + threadIdx.x;
    if (i >= NBOX) return;
    float k = ((keepmask[i >> 5] >> (i & 31)) & 1u) ? 1.0f : 0.0f;
    float4 v = make_float4(st[i] * k, en[i] * k, pk[i] * k, hh[i] * k);
    *(float4*)(out + (size_t)i * 4) = v;          // global_store_b128
}

// ---------------------------------------------------------------------------
// Workspace layout (≈ 388 KB):
//   [0    , 128K) uint64 items
//   [128K , 192K) float st      [192K , 256K) float en
//   [256K , 320K) float pk      [320K , 384K) float h
//   [384K , +2K ) u32 validmask [+2K  , +4K ) u32 keepmask
// ---------------------------------------------------------------------------
extern "C" void kernel_launch(void* const* d_in, const int* in_sizes, int n_in,
                              void* d_out, int out_size, void* d_ws, size_t ws_size,
                              hipStream_t stream) {
    (void)in_sizes; (void)n_in; (void)out_size; (void)ws_size;
    const float* out5 = (const float*)d_in[0];
    char* ws = (char*)d_ws;

    unsigned long long* items = (unsigned long long*)ws;
    float* st = (float*)(ws + (size_t)131072);
    float* en = st + NBOX;
    float* pk = en + NBOX;
    float* hh = pk + NBOX;
    unsigned* validmask = (unsigned*)(hh + NBOX);
    unsigned* keepmask  = validmask + NWORDS;
    float* out = (float*)d_out;

    const int TB = 256;
    const int GB = (NBOX + TB - 1) / TB;

    pack_kernel  <<<GB, TB, 0, stream>>>(out5, items, validmask);
    sort_kernel  <<<1, 1024, 0, stream>>>(items);
    gather_kernel<<<GB, TB, 0, stream>>>(out5, items, st, en, pk, hh, validmask);
    nms_kernel   <<<1, 1024, 0, stream>>>(st, en, pk, hh, validmask, keepmask);
    write_kernel <<<GB, TB, 0, stream>>>(st, en, pk, hh, keepmask, out);
}